// GATv2_38276748541989
// MI455X (gfx1250) — compile-verified
//
#include <hip/hip_runtime.h>
#include <hip/hip_bf16.h>
#include <math.h>

// Problem constants (match reference)
#define N_NODES  50000
#define N_EDGES  800000
#define N_GRAPHS 512
#define DIN      128
#define DH_      128
#define H1       2
#define F1       256          // H1*DH_
#define DOUT     128
#define NEG_SLOPE 0.2f
#define EPS_BN    1e-5f

typedef float v2f __attribute__((ext_vector_type(2)));
typedef float v8f __attribute__((ext_vector_type(8)));

__device__ __forceinline__ void atomicMaxF(float* addr, float val) {
    // works for mixed-sign floats when initialized to -inf
    if (val >= 0.0f) atomicMax((int*)addr, __float_as_int(val));
    else             atomicMin((unsigned int*)addr, __float_as_uint(val));
}

__global__ void fill_kernel(float* __restrict__ p, float v, long n) {
    long i = (long)blockIdx.x * blockDim.x + threadIdx.x;
    if (i < n) p[i] = v;
}

// ---------------------------------------------------------------------------
// fp32 WMMA GEMM: Y[n x NC] = X[n x K] @ W[K x NC], one 16x16 tile per wave.
// Uses V_WMMA_F32_16X16X4_F32 (exact fp32, matches reference numerics).
// X strip is staged into LDS with GLOBAL_LOAD_ASYNC_TO_LDS_B128 (ASYNCcnt),
// the CDNA5 async global->LDS path, then consumed via ds loads.
// A layout (ISA 7.12.2, 32-bit A 16x4): lane&15 = M, (lane>>4)*2+j = K.
// B layout (C/D-mirrored):              lane&15 = N, (lane>>4)*2+j = K.
// C/D layout: row = (lane>>4)*8 + vgpr, col = lane&15.
// ---------------------------------------------------------------------------
template<int K, int NW>
__global__ __launch_bounds__(NW * 32)
void gemm_wmma(const float* __restrict__ X, const float* __restrict__ W,
               float* __restrict__ Y) {
    constexpr int NC   = NW * 16;
    constexpr int LDSK = K + 4;            // pad to avoid bank conflicts
    __shared__ float sx[16 * LDSK];
    const int tid = threadIdx.x;
    const int rt  = blockIdx.x;            // 16-row strip index

    // async copy 16xK strip to LDS: one b128 (4 floats) per lane per iter
    constexpr int VECS = (16 * K) / 4;
    for (int idx = tid; idx < VECS; idx += NW * 32) {
        const int r = (idx * 4) / K, c = (idx * 4) % K;
        const unsigned ldsOff = (unsigned)(uintptr_t)(&sx[r * LDSK + c]);
        const float* gp = X + (size_t)(rt * 16 + r) * K + c;
        asm volatile("global_load_async_to_lds_b128 %0, %1, off"
                     :: "v"(ldsOff), "v"(gp) : "memory");
    }
    asm volatile("s_wait_asynccnt 0" ::: "memory");
    __syncthreads();

    const int lane = tid & 31;
    const int ct   = tid >> 5;             // column tile = wave id
    const int m    = lane & 15;            // M for A, N for B
    const int koff = (lane >> 4) * 2;

    v8f acc = {};
    #pragma unroll 4
    for (int kt = 0; kt < K / 4; ++kt) {
        const int kb = kt * 4 + koff;
        v2f a, b;
        a.x = sx[m * LDSK + kb];
        a.y = sx[m * LDSK + kb + 1];
        b.x = W[(size_t)kb       * NC + ct * 16 + m];
        b.y = W[(size_t)(kb + 1) * NC + ct * 16 + m];
        acc = __builtin_amdgcn_wmma_f32_16x16x4_f32(
                  false, a, false, b, (short)0, acc, false, false);
    }

    const int rbase = rt * 16 + (lane >> 4) * 8;
    const int col   = ct * 16 + m;
    #pragma unroll
    for (int v = 0; v < 8; ++v)
        Y[(size_t)(rbase + v) * NC + col] = acc[v];
}

// ---------------------------------------------------------------------------
// Edge score: e[edge,h] = sum_d leakyrelu(xl[s]+xr[d]) * att[h]; atomicMax emax
// One wave per edge (wave32). H=2 -> 16-lane head groups, H=1 -> full wave.
// ---------------------------------------------------------------------------
template<int H>
__global__ __launch_bounds__(256)
void edge_scores(const float* __restrict__ xl, const float* __restrict__ xr,
                 const float* __restrict__ att, const int* __restrict__ ei,
                 float* __restrict__ ebuf, float* __restrict__ emax) {
    constexpr int F  = H * DH_;
    constexpr int CH = F / 32;          // floats per lane
    constexpr int G  = 32 / H;          // lanes per head group
    const int  lane = threadIdx.x & 31;
    const long edge = (long)blockIdx.x * 8 + (threadIdx.x >> 5);
    const long Etot = (long)N_EDGES + N_NODES;
    if (edge >= Etot) return;

    int s, d;
    if (edge < N_EDGES) { s = ei[edge]; d = ei[N_EDGES + edge]; }
    else                { s = d = (int)(edge - N_EDGES); }

    const int f0   = lane * CH;
    const int head = f0 / DH_;
    const float* pl = xl  + (size_t)s * F + f0;
    const float* pr = xr  + (size_t)d * F + f0;
    const float* pa = att + head * DH_ + (f0 - head * DH_);

    float partial = 0.0f;
    #pragma unroll
    for (int j = 0; j < CH; j += 4) {
        float4 a4 = *reinterpret_cast<const float4*>(pl + j);
        float4 b4 = *reinterpret_cast<const float4*>(pr + j);
        float4 t4 = *reinterpret_cast<const float4*>(pa + j);
        float m0 = a4.x + b4.x; m0 = m0 > 0.f ? m0 : m0 * NEG_SLOPE; partial += m0 * t4.x;
        float m1 = a4.y + b4.y; m1 = m1 > 0.f ? m1 : m1 * NEG_SLOPE; partial += m1 * t4.y;
        float m2 = a4.z + b4.z; m2 = m2 > 0.f ? m2 : m2 * NEG_SLOPE; partial += m2 * t4.z;
        float m3 = a4.w + b4.w; m3 = m3 > 0.f ? m3 : m3 * NEG_SLOPE; partial += m3 * t4.w;
    }
    #pragma unroll
    for (int off = 1; off < G; off <<= 1)
        partial += __shfl_xor(partial, off, 32);

    if ((lane & (G - 1)) == 0) {
        const int h = lane / G;
        ebuf[edge * H + h] = partial;
        atomicMaxF(&emax[(size_t)d * H + h], partial);
    }
}

template<int H>
__global__ void edge_softmax_num(const int* __restrict__ ei, float* __restrict__ ebuf,
                                 const float* __restrict__ emax, float* __restrict__ denom) {
    const long Etot = (long)N_EDGES + N_NODES;
    long idx = (long)blockIdx.x * blockDim.x + threadIdx.x;
    if (idx >= Etot * H) return;
    const long edge = idx / H;
    const int  h    = (int)(idx % H);
    const int  d    = (edge < N_EDGES) ? ei[N_EDGES + edge] : (int)(edge - N_EDGES);
    const float v   = expf(ebuf[idx] - emax[(size_t)d * H + h]);
    ebuf[idx] = v;
    atomicAdd(&denom[(size_t)d * H + h], v);
}

template<int H>
__global__ void edge_aggregate(const float* __restrict__ xl, const float* __restrict__ ebuf,
                               const float* __restrict__ denom, const int* __restrict__ ei,
                               float* __restrict__ out) {
    constexpr int F = H * DH_;           // blockDim.x == F
    const long Etot = (long)N_EDGES + N_NODES;
    const int  f    = threadIdx.x;
    const int  head = f / DH_;
    for (long edge = blockIdx.x; edge < Etot; edge += gridDim.x) {
        int s, d;
        if (edge < N_EDGES) { s = ei[edge]; d = ei[N_EDGES + edge]; }
        else                { s = d = (int)(edge - N_EDGES); }
        const float alpha = ebuf[edge * H + head] /
                            (denom[(size_t)d * H + head] + 1e-16f);
        atomicAdd(&out[(size_t)d * F + f], xl[(size_t)s * F + f] * alpha);
    }
}

// bias + ELU in place, accumulate per-channel sum / sumsq (F1 channels)
__global__ __launch_bounds__(256)
void bias_elu_bnstats(float* __restrict__ h, const float* __restrict__ b,
                      float* __restrict__ bnsum, float* __restrict__ bnsq,
                      int rowsPer) {
    const int f = threadIdx.x;
    const float bias = b[f];
    float s = 0.0f, sq = 0.0f;
    const int n0 = blockIdx.x * rowsPer;
    const int n1 = min(n0 + rowsPer, N_NODES);
    for (int n = n0; n < n1; ++n) {
        float v = h[(size_t)n * F1 + f] + bias;
        v = v > 0.0f ? v : expm1f(v);          // ELU
        h[(size_t)n * F1 + f] = v;
        s += v; sq += v * v;
    }
    atomicAdd(&bnsum[f], s);
    atomicAdd(&bnsq[f], sq);
}

__global__ void bn_finalize(const float* __restrict__ bnsum, const float* __restrict__ bnsq,
                            const float* __restrict__ gamma, const float* __restrict__ beta,
                            float* __restrict__ scale, float* __restrict__ shift) {
    const int f = threadIdx.x;                 // 256 threads
    const float mu  = bnsum[f] / (float)N_NODES;
    const float var = bnsq[f] / (float)N_NODES - mu * mu;
    const float sc  = rsqrtf(var + EPS_BN) * gamma[f];
    scale[f] = sc;
    shift[f] = beta[f] - mu * sc;
}

__global__ void bn_apply(float* __restrict__ h, const float* __restrict__ scale,
                         const float* __restrict__ shift) {
    size_t i = (size_t)blockIdx.x * blockDim.x + threadIdx.x;
    if (i >= (size_t)N_NODES * F1) return;
    const int f = (int)(i & 255);              // F1 == 256
    h[i] = h[i] * scale[f] + shift[f];
}

__global__ void pool_sum(const float* __restrict__ h2, const int* __restrict__ batch,
                         float* __restrict__ pooled, float* __restrict__ cnt) {
    size_t i = (size_t)blockIdx.x * blockDim.x + threadIdx.x;
    if (i >= (size_t)N_NODES * DOUT) return;
    const int n = (int)(i >> 7);               // DOUT == 128
    const int f = (int)(i & 127);
    const int g = batch[n];
    atomicAdd(&pooled[(size_t)g * DOUT + f], h2[i]);
    if (f == 0) atomicAdd(&cnt[g], 1.0f);
}

__global__ void final_linear(const float* __restrict__ pooled, const float* __restrict__ cnt,
                             const float* __restrict__ b2, const float* __restrict__ Wlin,
                             const float* __restrict__ blin, float* __restrict__ out) {
    const int tid = blockIdx.x * blockDim.x + threadIdx.x;
    if (tid >= N_GRAPHS * 2) return;
    const int g = tid >> 1, c = tid & 1;
    const float cg  = cnt[g];
    const float inv = 1.0f / fmaxf(cg, 1.0f);
    float acc = blin[c];
    for (int f = 0; f < DOUT; ++f) {
        // per-node bias folded in: sum(out + b2) = sum + cnt*b2
        const float pm = (pooled[(size_t)g * DOUT + f] + cg * b2[f]) * inv;
        acc += pm * Wlin[f * 2 + c];
    }
    out[tid] = acc;
}

extern "C" void kernel_launch(void* const* d_in, const int* in_sizes, int n_in,
                              void* d_out, int out_size, void* d_ws, size_t ws_size,
                              hipStream_t stream) {
    const float* x     = (const float*)d_in[0];
    const int*   ei    = (const int*)  d_in[1];
    const int*   batch = (const int*)  d_in[2];
    const float* Wl1   = (const float*)d_in[3];
    const float* Wr1   = (const float*)d_in[4];
    const float* att1  = (const float*)d_in[5];
    const float* b1    = (const float*)d_in[6];
    const float* gamma = (const float*)d_in[7];
    const float* beta  = (const float*)d_in[8];
    const float* Wl2   = (const float*)d_in[9];
    const float* Wr2   = (const float*)d_in[10];
    const float* att2  = (const float*)d_in[11];
    const float* b2    = (const float*)d_in[12];
    const float* Wlin  = (const float*)d_in[13];
    const float* blin  = (const float*)d_in[14];
    float* out = (float*)d_out;

    // ---- workspace layout (floats) ----
    float* ws = (float*)d_ws;
    const size_t NF1  = (size_t)N_NODES * F1;          // 12.8M
    const size_t ETOT = (size_t)N_EDGES + N_NODES;     // 850k
    float* A    = ws;                  // xl1 [N,256]; later xl2 [N,128]
    float* B    = A + NF1;             // xr1 [N,256]; later xr2 [N,128] + out2
    float* C    = B + NF1;             // out1 -> h after bias/ELU/BN [N,256]
    float* D    = C + NF1;             // edge scores / exp [Etot*2]
    float* EM   = D + ETOT * 2;        // segment max  [N*2]
    float* DEN  = EM + (size_t)N_NODES * 2;  // segment sum [N*2]
    float* BN   = DEN + (size_t)N_NODES * 2; // 1024: sum|sq|scale|shift
    float* PO   = BN + 1024;                 // pooled [G,128]
    float* CNT  = PO + (size_t)N_GRAPHS * DOUT; // counts [G]
    float* out2 = B + (size_t)N_NODES * DOUT;   // second half of B

    const int RT = N_NODES / 16;       // 3125 row tiles (exact)
    const float NEG_INF = -__builtin_huge_valf();
    auto cdiv = [](long a, long b) { return (int)((a + b - 1) / b); };

    // ================= Layer 1 =================
    gemm_wmma<128, 16><<<RT, 512, 0, stream>>>(x, Wl1, A);
    gemm_wmma<128, 16><<<RT, 512, 0, stream>>>(x, Wr1, B);

    fill_kernel<<<cdiv(NF1, 256), 256, 0, stream>>>(C, 0.0f, (long)NF1);
    fill_kernel<<<cdiv(N_NODES * 2, 256), 256, 0, stream>>>(EM, NEG_INF, (long)N_NODES * 2);
    fill_kernel<<<cdiv(N_NODES * 2, 256), 256, 0, stream>>>(DEN, 0.0f, (long)N_NODES * 2);
    fill_kernel<<<4, 256, 0, stream>>>(BN, 0.0f, 1024);

    edge_scores<2><<<cdiv(ETOT, 8), 256, 0, stream>>>(A, B, att1, ei, D, EM);
    edge_softmax_num<2><<<cdiv(ETOT * 2, 256), 256, 0, stream>>>(ei, D, EM, DEN);
    edge_aggregate<2><<<4096, 256, 0, stream>>>(A, D, DEN, ei, C);

    bias_elu_bnstats<<<cdiv(N_NODES, 250), 256, 0, stream>>>(C, b1, BN, BN + 256, 250);
    bn_finalize<<<1, 256, 0, stream>>>(BN, BN + 256, gamma, beta, BN + 512, BN + 768);
    bn_apply<<<cdiv(NF1, 256), 256, 0, stream>>>(C, BN + 512, BN + 768);

    // ================= Layer 2 =================
    gemm_wmma<256, 8><<<RT, 256, 0, stream>>>(C, Wl2, A);
    gemm_wmma<256, 8><<<RT, 256, 0, stream>>>(C, Wr2, B);

    fill_kernel<<<cdiv((size_t)N_NODES * DOUT, 256), 256, 0, stream>>>(out2, 0.0f, (long)N_NODES * DOUT);
    fill_kernel<<<cdiv(N_NODES, 256), 256, 0, stream>>>(EM, NEG_INF, (long)N_NODES);
    fill_kernel<<<cdiv(N_NODES, 256), 256, 0, stream>>>(DEN, 0.0f, (long)N_NODES);

    edge_scores<1><<<cdiv(ETOT, 8), 256, 0, stream>>>(A, B, att2, ei, D, EM);
    edge_softmax_num<1><<<cdiv(ETOT, 256), 256, 0, stream>>>(ei, D, EM, DEN);
    edge_aggregate<1><<<4096, 128, 0, stream>>>(A, D, DEN, ei, out2);

    // ================= Pool + classifier =================
    fill_kernel<<<cdiv(N_GRAPHS * DOUT, 256), 256, 0, stream>>>(PO, 0.0f, (long)N_GRAPHS * DOUT);
    fill_kernel<<<2, 256, 0, stream>>>(CNT, 0.0f, N_GRAPHS);
    pool_sum<<<cdiv((size_t)N_NODES * DOUT, 256), 256, 0, stream>>>(out2, batch, PO, CNT);
    final_linear<<<cdiv(N_GRAPHS * 2, 64), 64, 0, stream>>>(PO, CNT, b2, Wlin, blin, out);
}